// ChunkwiseRetention_41575283425485
// MI455X (gfx1250) — compile-verified
//
#include <hip/hip_runtime.h>
#include <stdint.h>

// ---------------------------------------------------------------------------
// ChunkwiseRetention for MI455X (gfx1250), f32 WMMA (16x16x4) throughout.
// Kernel 1: qkv = x @ W + b          (16384 x 512 x 1536 GEMM)
// Kernel 2: chunk-parallel retention scan, superchunk T=128 (64 micro-chunks),
//           per-(batch, 64-col state slice) workgroup; state lives in LDS.
//           V slice staged into LDS with GLOBAL_LOAD_ASYNC_TO_LDS_B128,
//           overlapped with the Q K^T stage, fenced by s_wait_asynccnt.
// ---------------------------------------------------------------------------

typedef float v2f __attribute__((ext_vector_type(2)));
typedef float v8f __attribute__((ext_vector_type(8)));

#define WMMA_F32(a, b, c) \
  __builtin_amdgcn_wmma_f32_16x16x4_f32(false, (a), false, (b), (short)0, (c), false, false)

constexpr int Bk = 8;
constexpr int Sk = 2048;
constexpr int Dk = 512;      // d_in
constexpr int Hk = 512;      // hidden
constexpr int N3 = 1536;     // 3*H
constexpr float GAMMA_F = 0.984375f;

constexpr int MT   = 64;       // state-column tile per workgroup
constexpr int T    = 128;      // superchunk positions (64 micro-chunks of 2)
constexpr int NSC  = Sk / T;   // 16 sequential superchunks
constexpr int SSTR = MT + 1;   // LDS row stride for S
constexpr int ASTR = T + 1;    // LDS row stride for A / Acoef
constexpr int VSTR = MT + 4;   // LDS row stride for V slice (16B-aligned rows)

// Async copy 16B from global to LDS (ASYNCcnt-tracked). vdst = LDS byte
// offset (low 32 bits of the generic LDS pointer), vaddr = 64-bit address.
__device__ __forceinline__ void async_b128_to_lds(float* l, const float* g) {
  const uint32_t ldsOff = (uint32_t)(uintptr_t)l;
  asm volatile("global_load_async_to_lds_b128 %0, %1, off"
               :: "v"(ldsOff), "v"(g)
               : "memory");
}
__device__ __forceinline__ void wait_async_zero() {
  asm volatile("s_wait_asynccnt 0x0" ::: "memory");
}

// ---------------------------------------------------------------------------
// Kernel 1: qkv = x @ W + bias.  Block = 256 thr (8 waves); block tile 256x64;
// each wave owns two 16-row strips (M, M+128) x 4 16x16 accumulators.
// ---------------------------------------------------------------------------
__global__ void qkv_gemm_kernel(const float* __restrict__ x,
                                const float* __restrict__ W,
                                const float* __restrict__ bias,
                                float* __restrict__ qkv) {
  const int lane = threadIdx.x & 31;
  const int wave = threadIdx.x >> 5;
  const int half = lane >> 4;   // selects K pair per f32-WMMA A/B layout
  const int l16  = lane & 15;

  const int mBase = blockIdx.x * 256 + wave * 16;
  const int nBase = blockIdx.y * 64;

  const float* xrow0 = x + (size_t)(mBase + l16) * Dk;
  const float* xrow1 = x + (size_t)(mBase + 128 + l16) * Dk;

  v8f acc[2][4] = {};
  for (int k = 0; k < Dk; k += 4) {
    const int k0 = k + half * 2;
    v2f a0, a1;
    a0.x = xrow0[k0];
    a0.y = xrow0[k0 + 1];
    a1.x = xrow1[k0];
    a1.y = xrow1[k0 + 1];
    // Branchless near-cache prefetch of the upcoming A-fragment columns
    // (locality 3 -> WGP scope: pull into all cache levels). Offset wraps
    // inside the row so the address is always valid.
    {
      const int kpf = (k + 16) & (Dk - 1);
      __builtin_prefetch(xrow0 + kpf, 0, 3);   // global_prefetch_b8, WGP scope
      __builtin_prefetch(xrow1 + kpf, 0, 3);
    }
#pragma unroll
    for (int tc = 0; tc < 4; ++tc) {
      const int n = nBase + tc * 16 + l16;
      v2f bb;
      bb.x = W[(size_t)k0 * N3 + n];
      bb.y = W[(size_t)(k0 + 1) * N3 + n];
      acc[0][tc] = WMMA_F32(a0, bb, acc[0][tc]);
      acc[1][tc] = WMMA_F32(a1, bb, acc[1][tc]);
    }
  }
#pragma unroll
  for (int r = 0; r < 2; ++r) {
#pragma unroll
    for (int tc = 0; tc < 4; ++tc) {
      const int n  = nBase + tc * 16 + l16;
      const float bv = bias[n];
#pragma unroll
      for (int v = 0; v < 8; ++v) {
        const int m = mBase + r * 128 + (half ? v + 8 : v);
        qkv[(size_t)m * N3 + n] = acc[r][tc][v] + bv;
      }
    }
  }
}

// ---------------------------------------------------------------------------
// Kernel 2: retention scan.  grid = (8 batches, 8 column tiles of 64).
// LDS: S 512x65 (130KB) + A 128x129 + Acoef 128x129 (~129KB) + V 128x68 (34KB).
// Per superchunk:
//   (v) async-stage V slice into LDS (overlaps with (a))
//   (a) A = Q K^T (128x128, K=512)                 -> LDS
//   (b) Acoef: decay mask (gq*gc^(u-1-u')) on strict-lower micro-blocks,
//       normalized 2x2 attention (inv_sqrt_h * Dsub / denom) on the diagonal
//   (c) out = Acoef @ Vlds  +  (gq*gc^u * Q) @ S_slice
//   (e) S = gc^64 * S + (gc^(63-u) K)^T @ Vlds
// ---------------------------------------------------------------------------
__global__ void retention_scan_kernel(const float* __restrict__ qkv,
                                      float* __restrict__ out) {
  extern __shared__ float lds[];
  float* Sst   = lds;                        // 512 * SSTR
  float* Amat  = Sst + Hk * SSTR;            // T * ASTR
  float* Acoef = Amat + T * ASTR;            // T * ASTR
  float* Vlds  = Acoef + T * ASTR;           // T * VSTR
  float* dec   = Vlds + T * VSTR;            // 64 entries: gc^i

  const int lane = threadIdx.x & 31;
  const int wave = threadIdx.x >> 5;
  const int half = lane >> 4;
  const int l16  = lane & 15;
  const int b     = blockIdx.x;
  const int mBase = blockIdx.y * MT;

  const float g   = GAMMA_F;
  const float gc  = g * g;   // gamma^CHUNK
  const float gq  = gc;      // gamma^(COUNTER+1)
  const float inv_sqrt_h = 0.044194173824159216f;   // 1/sqrt(512)
  const float r1 = 1.f + g;
  const float r2 = 1.f + g + g * g;
  const float ds00 = rsqrtf(r1);
  const float ds10 = g * rsqrtf(r1);
  const float ds11 = rsqrtf(r2);          // Dsub[0][1] == 0

  for (int i = threadIdx.x; i < 64; i += blockDim.x) dec[i] = powf(gc, (float)i);
  for (int i = threadIdx.x; i < Hk * SSTR; i += blockDim.x) Sst[i] = 0.f;
  __syncthreads();
  const float gc64 = dec[63] * gc;        // gc^64

  const float* base = qkv + (size_t)b * Sk * N3;

  for (int sc = 0; sc < NSC; ++sc) {
    const float* Qg = base + (size_t)sc * T * N3;   // q at row offset 0
    const float* Kg = Qg + Hk;                      // k at +512
    const float* Vg = Qg + 2 * Hk + mBase;          // v column slice

    // ---- (v) async-stage the V slice (128 x 64 f32) into LDS -------------
    // 2048 x 16B chunks; 256 threads x 8 issues. Overlaps with stage (a).
    {
#pragma unroll
      for (int it = 0; it < 8; ++it) {
        const int i   = threadIdx.x + it * 256;   // chunk id
        const int row = i >> 4;                   // 16 chunks per 64-f32 row
        const int c4  = (i & 15) * 4;             // float column
        async_b128_to_lds(Vlds + row * VSTR + c4, Vg + (size_t)row * N3 + c4);
      }
    }

    // ---- (a) A = Q K^T ---------------------------------------------------
    {
      const float* qrow = Qg + (size_t)(wave * 16 + l16) * N3;
      for (int tc = 0; tc < 8; ++tc) {
        const float* krow = Kg + (size_t)(tc * 16 + l16) * N3;
        v8f acc = {};
        for (int k = 0; k < Hk; k += 4) {
          const int k0 = k + half * 2;
          v2f a;  a.x  = qrow[k0]; a.y  = qrow[k0 + 1];
          v2f bb; bb.x = krow[k0]; bb.y = krow[k0 + 1];
          acc = WMMA_F32(a, bb, acc);
        }
#pragma unroll
        for (int v = 0; v < 8; ++v) {
          const int t  = wave * 16 + (half ? v + 8 : v);
          const int tp = tc * 16 + l16;
          Amat[t * ASTR + tp] = acc[v];
        }
      }
    }
    __syncthreads();

    // ---- (b) build coefficient matrix ------------------------------------
    for (int idx = threadIdx.x; idx < T * T; idx += blockDim.x) {
      const int t  = idx >> 7;
      const int tp = idx & 127;
      const int u  = t >> 1;
      const int up = tp >> 1;
      const float val = Amat[t * ASTR + tp];
      float o;
      if (up > u) {
        o = 0.f;
      } else if (up < u) {
        o = gq * dec[u - 1 - up] * val;          // gv = gamma^0 = 1
      } else {
        const int c = t & 1, d = tp & 1;
        const float dsub = (c == 0) ? (d == 0 ? ds00 : 0.f)
                                    : (d == 0 ? ds10 : ds11);
        const float att = val * inv_sqrt_h * dsub;
        const float a0 = Amat[(2 * u) * ASTR + tp]     * inv_sqrt_h * (d == 0 ? ds00 : 0.f);
        const float a1 = Amat[(2 * u + 1) * ASTR + tp] * inv_sqrt_h * (d == 0 ? ds10 : ds11);
        const float denom = fmaxf(fabsf(a0 + a1), 1.f);
        o = att / denom;
      }
      Acoef[t * ASTR + tp] = o;
    }
    wait_async_zero();      // V slice in LDS (each thread's own issues)
    __syncthreads();        // ... and everyone else's

    // ---- (c) out = Acoef @ Vlds + (scaled Q) @ S -------------------------
    {
      const int   m        = wave * 16 + l16;        // A-fragment row
      const float rowScale = gq * dec[m >> 1];       // gq * gc^u
      const float* qrow = Qg + (size_t)m * N3;
      const float* arow = Acoef + m * ASTR;
#pragma unroll
      for (int tc = 0; tc < 4; ++tc) {
        v8f acc = {};
        for (int k = 0; k < T; k += 4) {             // Acoef @ V, K = 128
          const int k0 = k + half * 2;
          v2f a;  a.x = arow[k0]; a.y = arow[k0 + 1];
          v2f bb;
          bb.x = Vlds[k0 * VSTR + tc * 16 + l16];
          bb.y = Vlds[(k0 + 1) * VSTR + tc * 16 + l16];
          acc = WMMA_F32(a, bb, acc);
        }
        for (int k = 0; k < Hk; k += 4) {            // Qs @ S, K = 512
          const int k0 = k + half * 2;
          v2f a;
          a.x = qrow[k0] * rowScale;
          a.y = qrow[k0 + 1] * rowScale;
          v2f bb;
          bb.x = Sst[k0 * SSTR + tc * 16 + l16];
          bb.y = Sst[(k0 + 1) * SSTR + tc * 16 + l16];
          acc = WMMA_F32(a, bb, acc);
        }
#pragma unroll
        for (int v = 0; v < 8; ++v) {
          const int t = wave * 16 + (half ? v + 8 : v);
          out[((size_t)b * Sk + (size_t)sc * T + t) * Hk + mBase + tc * 16 + l16] = acc[v];
        }
      }
    }
    __syncthreads();

    // ---- (e) S = gc^64 * S + K_d^T @ Vlds --------------------------------
    {
#pragma unroll
      for (int trr = 0; trr < 4; ++trr) {
        const int hBase = wave * 64 + trr * 16;
#pragma unroll
        for (int tc = 0; tc < 4; ++tc) {
          v8f acc;
#pragma unroll
          for (int v = 0; v < 8; ++v)
            acc[v] = gc64 * Sst[(hBase + (half ? v + 8 : v)) * SSTR + tc * 16 + l16];
          for (int k = 0; k < T; k += 4) {
            const int k0 = k + half * 2;           // k0 even -> k0,k0+1 share micro-chunk
            const float scale = dec[63 - (k0 >> 1)];
            v2f a;
            a.x = scale * Kg[(size_t)k0 * N3 + hBase + l16];
            a.y = scale * Kg[(size_t)(k0 + 1) * N3 + hBase + l16];
            v2f bb;
            bb.x = Vlds[k0 * VSTR + tc * 16 + l16];
            bb.y = Vlds[(k0 + 1) * VSTR + tc * 16 + l16];
            acc = WMMA_F32(a, bb, acc);
          }
#pragma unroll
          for (int v = 0; v < 8; ++v)
            Sst[(hBase + (half ? v + 8 : v)) * SSTR + tc * 16 + l16] = acc[v];
        }
      }
    }
    __syncthreads();
  }
}

// ---------------------------------------------------------------------------
extern "C" void kernel_launch(void* const* d_in, const int* in_sizes, int n_in,
                              void* d_out, int out_size, void* d_ws, size_t ws_size,
                              hipStream_t stream) {
  const float* x    = (const float*)d_in[0];   // (8, 2048, 512)
  const float* W    = (const float*)d_in[1];   // (512, 1536)
  const float* bias = (const float*)d_in[2];   // (1536,)
  float* out = (float*)d_out;                  // (8, 2048, 512)
  float* qkv = (float*)d_ws;                   // 8*2048*1536 f32 = 96 MB scratch

  // Phase 1: qkv GEMM. Grid covers M=16384 (tiles of 256), N=1536 (tiles of 64).
  dim3 g1((Bk * Sk) / 256, N3 / 64);
  qkv_gemm_kernel<<<g1, 256, 0, stream>>>(x, W, bias, qkv);

  // Phase 2: retention scan. One WG per (batch, 64-wide state-column slice).
  const size_t ldsBytes =
      (size_t)(Hk * SSTR + 2 * T * ASTR + T * VSTR + 64) * sizeof(float);  // ~300 KB
  dim3 g2(Bk, Hk / MT);
  retention_scan_kernel<<<g2, 256, ldsBytes, stream>>>(qkv, out);
}